// ModelEuCoHT_2963527434891
// MI455X (gfx1250) — compile-verified
//
#include <hip/hip_runtime.h>

// Problem constants (from reference)
#define NA_N 100000
#define NR_N 200000
#define E_N  600000
#define EL_N 500000
#define H_N  128
#define NEG_SLOPE 0.2f
#define BN_EPS 1e-5f

typedef float v2f __attribute__((ext_vector_type(2)));
typedef float v8f __attribute__((ext_vector_type(8)));

// ---------------------------------------------------------------------------
// utility fills (grid-stride, deterministic)
// ---------------------------------------------------------------------------
__launch_bounds__(256)
__global__ void fill_f32(float* __restrict__ p, float v, int n) {
  int i = blockIdx.x * blockDim.x + threadIdx.x;
  int st = gridDim.x * blockDim.x;
  for (; i < n; i += st) p[i] = v;
}

__launch_bounds__(256)
__global__ void fill_i32(int* __restrict__ p, int v, int n) {
  int i = blockIdx.x * blockDim.x + threadIdx.x;
  int st = gridDim.x * blockDim.x;
  for (; i < n; i += st) p[i] = v;
}

// ---------------------------------------------------------------------------
// Y = A @ W (+ A + bias when RESID) for H=128, via V_WMMA_F32_16X16X4_F32.
// Block: 256 threads = 8 waves; 64-row A tile in LDS (stride 132 floats ->
// bank-conflict-free: bank = (4*m + k) % 64). Wave w: row-tile (w&3),
// col-group (w>>2) covering 4 col-tiles of 16. K=128 -> 32 chained WMMA/tile.
// RESID is a template flag so the epilogue is straight-line code, and full
// tiles (r0+64 <= N) take an unguarded store path.
// ---------------------------------------------------------------------------
template <bool RESID>
__launch_bounds__(256)
__global__ void gemm_h128_wmma(const float* __restrict__ A,
                               const float* __restrict__ W,
                               const float* __restrict__ bias,
                               float* __restrict__ Y,
                               int N) {
  __shared__ __align__(16) float As[64 * 132];
  const int r0  = blockIdx.x * 64;
  const int tid = threadIdx.x;

  // Stage 64x128 f32 tile (zero-padded past N) — float4 vectorized, coalesced.
  for (int v = tid; v < 64 * 32; v += 256) {
    const int row = v >> 5;            // 32 float4 per row
    const int c4  = (v & 31) << 2;
    float4 val = make_float4(0.f, 0.f, 0.f, 0.f);
    if (r0 + row < N)
      val = *(const float4*)(A + (size_t)(r0 + row) * H_N + c4);
    *(float4*)(&As[row * 132 + c4]) = val;   // 528B row stride, 16B aligned
  }
  __syncthreads();

  const int wv   = tid >> 5;
  const int lane = tid & 31;
  const int l16  = lane & 15;
  const int hi   = lane >> 4;          // 0: K{0,1}, 1: K{2,3} (A & B frags)
  const int rowBase = (wv & 3) * 16;   // row tile
  const int cg      = (wv >> 2) * 4;   // first col tile of this wave
  const bool fullTile = (r0 + 64) <= N;

  const float* arow = &As[(rowBase + l16) * 132];

  for (int ct = 0; ct < 4; ++ct) {
    const int col = (cg + ct) * 16 + l16;
    v8f acc = {0.f, 0.f, 0.f, 0.f, 0.f, 0.f, 0.f, 0.f};
    #pragma unroll 8
    for (int k = 0; k < 128; k += 4) {
      const int ka = k + (hi << 1);
      v2f a, b;
      a.x = arow[ka];
      a.y = arow[ka + 1];
      b.x = W[ka * H_N + col];
      b.y = W[(ka + 1) * H_N + col];
      acc = __builtin_amdgcn_wmma_f32_16x16x4_f32(
          /*neg_a=*/false, a, /*neg_b=*/false, b,
          /*c_mod=*/(short)0, acc, /*reuse_a=*/false, /*reuse_b=*/false);
    }
    const float bv = RESID ? bias[col] : 0.f;      // loop-invariant
    float* ybase = Y + (size_t)(r0 + rowBase + hi * 8) * H_N + col;
    const float* xbase = &As[(rowBase + hi * 8) * 132 + col];
    // D layout: VGPR r -> M = rowBase + 8*hi + r, N = col
    if (fullTile) {
      #pragma unroll
      for (int r = 0; r < 8; ++r) {
        float v = acc[r];
        if (RESID) v += xbase[r * 132] + bv;       // residual x + x@W + b
        ybase[(size_t)r * H_N] = v;
      }
    } else {
      const int grow0 = r0 + rowBase + hi * 8;
      #pragma unroll
      for (int r = 0; r < 8; ++r) {
        if (grow0 + r < N) {
          float v = acc[r];
          if (RESID) v += xbase[r * 132] + bv;
          ybase[(size_t)r * H_N] = v;
        }
      }
    }
  }
}

// ---------------------------------------------------------------------------
// order-preserving float<->int maps for atomic segment-max
// ---------------------------------------------------------------------------
__device__ __forceinline__ int f32_to_ordered(float x) {
  int i = __float_as_int(x);
  return (i >= 0) ? i : (i ^ 0x7fffffff);
}
__device__ __forceinline__ float ordered_to_f32(int o) {
  return __int_as_float((o >= 0) ? o : (o ^ 0x7fffffff));
}

// Pass A: e_j = leaky_relu(xl[s]+xr[d]) . a ; atomic segment max on dst.
// One wave per edge, lane = 4 contiguous features (float4, coalesced).
__launch_bounds__(256)
__global__ void edge_scores(const float* __restrict__ xl, const float* __restrict__ xr,
                            const int* __restrict__ src, const int* __restrict__ dst,
                            const float* __restrict__ att,
                            float* __restrict__ ebuf, int* __restrict__ mbuf, int ne) {
  const int j = blockIdx.x * 8 + (threadIdx.x >> 5);
  if (j >= ne) return;
  const int lane = threadIdx.x & 31;
  const int s = src[j], d = dst[j];
  const float4 a = *(const float4*)(att + lane * 4);
  const float4 u = *(const float4*)(xl + (size_t)s * H_N + lane * 4);
  const float4 v = *(const float4*)(xr + (size_t)d * H_N + lane * 4);
  float p = 0.f, t;
  t = u.x + v.x; t = (t > 0.f) ? t : NEG_SLOPE * t; p += t * a.x;
  t = u.y + v.y; t = (t > 0.f) ? t : NEG_SLOPE * t; p += t * a.y;
  t = u.z + v.z; t = (t > 0.f) ? t : NEG_SLOPE * t; p += t * a.z;
  t = u.w + v.w; t = (t > 0.f) ? t : NEG_SLOPE * t; p += t * a.w;
  #pragma unroll
  for (int off = 16; off > 0; off >>= 1) p += __shfl_xor(p, off, 32);
  if (lane == 0) {
    ebuf[j] = p;
    atomicMax(&mbuf[d], f32_to_ordered(p));
  }
}

// Pass B: ex = exp(e - m[d]); den[d] += ex (one thread per edge)
__launch_bounds__(256)
__global__ void edge_exp(float* __restrict__ ebuf, const int* __restrict__ dst,
                         const int* __restrict__ mbuf, float* __restrict__ den, int ne) {
  const int j = blockIdx.x * blockDim.x + threadIdx.x;
  if (j >= ne) return;
  const int d = dst[j];
  const float m = ordered_to_f32(mbuf[d]);
  const float ex = __expf(ebuf[j] - m);
  ebuf[j] = ex;
  atomicAdd(&den[d], ex);
}

// Pass C: out[d] += (ex/den[d]) * xl[s]  (wave per edge, f32 atomics at L2)
__launch_bounds__(256)
__global__ void edge_scatter(const float* __restrict__ ebuf, const float* __restrict__ xl,
                             const int* __restrict__ src, const int* __restrict__ dst,
                             const float* __restrict__ den, float* __restrict__ out, int ne) {
  const int j = blockIdx.x * 8 + (threadIdx.x >> 5);
  if (j >= ne) return;
  const int lane = threadIdx.x & 31;
  const int s = src[j], d = dst[j];
  const float alpha = ebuf[j] / den[d];
  const float4 u = *(const float4*)(xl + (size_t)s * H_N + lane * 4);
  float* o = out + (size_t)d * H_N + lane * 4;
  atomicAdd(o + 0, alpha * u.x);
  atomicAdd(o + 1, alpha * u.y);
  atomicAdd(o + 2, alpha * u.z);
  atomicAdd(o + 3, alpha * u.w);
}

// x[row, col] += b1[col] (+ b2[col])
__launch_bounds__(256)
__global__ void add_bias2(float* __restrict__ x, const float* __restrict__ b1,
                          const float* __restrict__ b2, int n) {
  const int total = n * H_N;
  int i = blockIdx.x * blockDim.x + threadIdx.x;
  const int st = gridDim.x * blockDim.x;
  for (; i < total; i += st) {
    const int col = i & (H_N - 1);
    float v = x[i] + b1[col];
    if (b2) v += b2[col];
    x[i] = v;
  }
}

// BN stats: per-column sum / sumsq into stats[0:128] / stats[128:256]
// (grid stride multiple of 128 => fixed column per thread)
__launch_bounds__(256)
__global__ void bn_stats(const float* __restrict__ x, float* __restrict__ stats, int n) {
  __shared__ float ssum[256], ssq[256];
  const int tid = threadIdx.x;
  const size_t total = (size_t)n * H_N;
  float s = 0.f, q = 0.f;
  for (size_t i = (size_t)blockIdx.x * 256 + tid; i < total; i += (size_t)gridDim.x * 256) {
    const float v = x[i];
    s += v; q += v * v;
  }
  ssum[tid] = s; ssq[tid] = q;
  __syncthreads();
  if (tid < 128) {
    atomicAdd(&stats[tid],       ssum[tid] + ssum[tid + 128]);
    atomicAdd(&stats[128 + tid], ssq[tid]  + ssq[tid + 128]);
  }
}

// BN apply (+ optional ReLU), writes to separate dest
__launch_bounds__(256)
__global__ void bn_apply(const float* __restrict__ x, const float* __restrict__ stats,
                         const float* __restrict__ g, const float* __restrict__ b,
                         float* __restrict__ y, int n, int relu) {
  const float inv_n = 1.f / (float)n;
  const int total = n * H_N;
  int i = blockIdx.x * blockDim.x + threadIdx.x;
  const int st = gridDim.x * blockDim.x;
  for (; i < total; i += st) {
    const int col = i & (H_N - 1);
    const float mean = stats[col] * inv_n;
    const float var  = stats[128 + col] * inv_n - mean * mean;
    float v = g[col] * (x[i] - mean) * rsqrtf(var + BN_EPS) + b[col];
    if (relu) v = fmaxf(v, 0.f);
    y[i] = v;
  }
}

// Final link predictor: out[j] = dot(z[i0[j]], z[i1[j]]) (wave per edge)
__launch_bounds__(256)
__global__ void edge_dot(const float* __restrict__ z, const int* __restrict__ i0,
                         const int* __restrict__ i1, float* __restrict__ out, int ne) {
  const int j = blockIdx.x * 8 + (threadIdx.x >> 5);
  if (j >= ne) return;
  const int lane = threadIdx.x & 31;
  const float4 a = *(const float4*)(z + (size_t)i0[j] * H_N + lane * 4);
  const float4 b = *(const float4*)(z + (size_t)i1[j] * H_N + lane * 4);
  float p = a.x * b.x + a.y * b.y + a.z * b.z + a.w * b.w;
  #pragma unroll
  for (int off = 16; off > 0; off >>= 1) p += __shfl_xor(p, off, 32);
  if (lane == 0) out[j] = p;
}

// ---------------------------------------------------------------------------
extern "C" void kernel_launch(void* const* d_in, const int* in_sizes, int n_in,
                              void* d_out, int out_size, void* d_ws, size_t ws_size,
                              hipStream_t stream) {
  const float* x_author  = (const float*)d_in[0];
  const float* x_article = (const float*)d_in[1];
  const float* Wl        = (const float*)d_in[2];   // [L,3,H,H]
  const float* Wr        = (const float*)d_in[3];
  const float* att       = (const float*)d_in[4];   // [L,3,H]
  const float* b_conv    = (const float*)d_in[5];   // [L,3,H]
  const float* lin_W     = (const float*)d_in[6];   // [L,2,H,H]
  const float* lin_b     = (const float*)d_in[7];
  const float* bn_g      = (const float*)d_in[8];
  const float* bn_b      = (const float*)d_in[9];
  const int* ei_pub = (const int*)d_in[10];         // [2,E] src then dst
  const int* ei_co  = (const int*)d_in[11];
  const int* ei_rev = (const int*)d_in[12];
  const int* eli    = (const int*)d_in[13];         // [2,EL]
  float* out = (float*)d_out;

  // workspace carve-up (f32)
  float* ws   = (float*)d_ws;
  float* ha   = ws;                                 // NA*H : layer-0 author out
  float* hr   = ha + (size_t)NA_N * H_N;            // NR*H : layer-0 article out
  float* na   = hr + (size_t)NR_N * H_N;            // NA*H : author accumulator
  float* nr   = na + (size_t)NA_N * H_N;            // NR*H : article accumulator
  float* xl   = nr + (size_t)NR_N * H_N;            // NR*H : src projection
  float* xr   = xl + (size_t)NR_N * H_N;            // NR*H : dst projection
  float* ebuf = xr + (size_t)NR_N * H_N;            // E
  float* den  = ebuf + E_N;                         // NR (max Nd)
  int*   mbuf = (int*)(den + NR_N);                 // NR
  float* stats = (float*)(mbuf + NR_N);             // 256

  const size_t HH = (size_t)H_N * H_N;
  const int FILLB = 1024;

  auto gemmPlain = [&](const float* A, const float* W, float* Y, int N) {
    gemm_h128_wmma<false><<<dim3((N + 63) / 64), 256, 0, stream>>>(A, W, nullptr, Y, N);
  };
  auto gemmResid = [&](const float* A, const float* W, const float* bias,
                       float* Y, int N) {
    gemm_h128_wmma<true><<<dim3((N + 63) / 64), 256, 0, stream>>>(A, W, bias, Y, N);
  };

  auto runGat = [&](const float* xs, int Ns, const float* xd, int Nd,
                    const int* ei, int layer, int et, float* accum) {
    const float* Wl_p = Wl + ((size_t)layer * 3 + et) * HH;
    const float* Wr_p = Wr + ((size_t)layer * 3 + et) * HH;
    const float* a_p  = att + ((size_t)layer * 3 + et) * H_N;
    gemmPlain(xs, Wl_p, xl, Ns);
    gemmPlain(xd, Wr_p, xr, Nd);
    fill_i32<<<256, 256, 0, stream>>>(mbuf, (int)0x80000000, Nd);
    fill_f32<<<256, 256, 0, stream>>>(den, 0.f, Nd);
    const int eb = (E_N + 7) / 8;
    edge_scores<<<eb, 256, 0, stream>>>(xl, xr, ei, ei + E_N, a_p, ebuf, mbuf, E_N);
    edge_exp<<<(E_N + 255) / 256, 256, 0, stream>>>(ebuf, ei + E_N, mbuf, den, E_N);
    edge_scatter<<<eb, 256, 0, stream>>>(ebuf, xl, ei, ei + E_N, den, accum, E_N);
  };

  // ------------------------- Layer 0 -------------------------
  fill_f32<<<FILLB, 256, 0, stream>>>(nr, 0.f, NR_N * H_N);
  fill_f32<<<FILLB, 256, 0, stream>>>(na, 0.f, NA_N * H_N);
  runGat(x_author,  NA_N, x_article, NR_N, ei_pub, 0, 0, nr);
  runGat(x_author,  NA_N, x_author,  NA_N, ei_co,  0, 1, na);
  runGat(x_article, NR_N, x_author,  NA_N, ei_rev, 0, 2, na);
  add_bias2<<<FILLB, 256, 0, stream>>>(nr, b_conv + 0 * H_N, nullptr, NR_N);
  add_bias2<<<FILLB, 256, 0, stream>>>(na, b_conv + 1 * H_N, b_conv + 2 * H_N, NA_N);
  // residual linear (in-place safe: block reads own rows into LDS before store)
  gemmResid(nr, lin_W + 0 * HH, lin_b + 0 * H_N, nr, NR_N);  // article
  gemmResid(na, lin_W + 1 * HH, lin_b + 1 * H_N, na, NA_N);  // author
  // BN + ReLU -> hr / ha
  fill_f32<<<1, 256, 0, stream>>>(stats, 0.f, 256);
  bn_stats<<<512, 256, 0, stream>>>(nr, stats, NR_N);
  bn_apply<<<FILLB, 256, 0, stream>>>(nr, stats, bn_g + 0 * H_N, bn_b + 0 * H_N, hr, NR_N, 1);
  fill_f32<<<1, 256, 0, stream>>>(stats, 0.f, 256);
  bn_stats<<<512, 256, 0, stream>>>(na, stats, NA_N);
  bn_apply<<<FILLB, 256, 0, stream>>>(na, stats, bn_g + 1 * H_N, bn_b + 1 * H_N, ha, NA_N, 1);

  // ------------------------- Layer 1 (author branch only; article output is
  // dead code in the reference: _encode returns ha) -------------------------
  fill_f32<<<FILLB, 256, 0, stream>>>(na, 0.f, NA_N * H_N);
  runGat(ha, NA_N, ha, NA_N, ei_co,  1, 1, na);
  runGat(hr, NR_N, ha, NA_N, ei_rev, 1, 2, na);
  add_bias2<<<FILLB, 256, 0, stream>>>(na, b_conv + (3 + 1) * H_N, b_conv + (3 + 2) * H_N, NA_N);
  gemmResid(na, lin_W + (2 + 1) * HH, lin_b + (2 + 1) * H_N, na, NA_N);
  fill_f32<<<1, 256, 0, stream>>>(stats, 0.f, 256);
  bn_stats<<<512, 256, 0, stream>>>(na, stats, NA_N);
  bn_apply<<<FILLB, 256, 0, stream>>>(na, stats, bn_g + (2 + 1) * H_N, bn_b + (2 + 1) * H_N,
                                      xl /* z */, NA_N, 0);  // no ReLU on last layer

  // ------------------------- link prediction -------------------------
  edge_dot<<<(EL_N + 7) / 8, 256, 0, stream>>>(xl, eli, eli + EL_N, out, EL_N);
}